// Jitter_22514218565654
// MI455X (gfx1250) — compile-verified
//
#include <hip/hip_runtime.h>
#include <stdint.h>

// Jitter: per-16x16-block shifted bilinear gather.
//   x:    [B,3,512,512] f32
//   flow: [B,2,32,32]   f32   (uniform per output block)
//   out:  [B,3,512,512] f32
//
// Key identity: ix = x + dx, dx = (flow*2-1)*0.5*(W-1) constant per block
// => integer shift + block-uniform bilinear weights, zero padding OOB.
// Data path: CDNA5 async global->LDS gather (ASYNCcnt), NT output stores.

namespace {
constexpr int IMG_H   = 512;
constexpr int IMG_W   = 512;
constexpr int BS      = 16;          // jitter block size (= thread block 16x16)
constexpr int NCH     = 3;
constexpr int TILE    = BS + 1;      // 17 : +1 ring for bilinear taps
constexpr int TELEMS  = TILE * TILE; // 289 per channel
constexpr int TOTAL   = NCH * TELEMS;// 867 floats of LDS tile
constexpr int NBX     = IMG_W / BS;  // 32
constexpr int NBY     = IMG_H / BS;  // 32
}

__global__ __launch_bounds__(BS * BS)
void jitter_tile_kernel(const float* __restrict__ x,
                        const float* __restrict__ flow,
                        float* __restrict__ out)
{
    __shared__ float tile[TOTAL];

    const int bx  = blockIdx.x;            // tile col  0..31
    const int by  = blockIdx.y;            // tile row  0..31
    const int b   = blockIdx.z;            // batch
    const int tx  = threadIdx.x;           // 0..15
    const int ty  = threadIdx.y;           // 0..15
    const int tid = ty * BS + tx;          // 0..255

    // ---- block-uniform flow -> integer shift + fractional weights ----
    // (addresses depend only on blockIdx -> compiler scalarizes to s_load)
    const float fx = flow[(b * 2 + 0) * (NBY * NBX) + by * NBX + bx];
    const float fy = flow[(b * 2 + 1) * (NBY * NBX) + by * NBX + bx];
    const float dx = (fx * 2.0f - 1.0f) * (0.5f * (float)(IMG_W - 1)); // PROB=1
    const float dy = (fy * 2.0f - 1.0f) * (0.5f * (float)(IMG_H - 1));
    const float fdx = floorf(dx);
    const float fdy = floorf(dy);
    const int   ix0 = (int)fdx;
    const int   iy0 = (int)fdy;
    const float wx  = dx - fdx;
    const float wy  = dy - fdy;
    const float w00 = (1.0f - wx) * (1.0f - wy);
    const float w01 = wx * (1.0f - wy);
    const float w10 = (1.0f - wx) * wy;
    const float w11 = wx * wy;

    const int gx0 = bx * BS + ix0;         // image-space origin of 17x17 tile
    const int gy0 = by * BS + iy0;

    // ---- single gather pass: each LDS slot written exactly once ----
    // Valid texel  -> GLOBAL_LOAD_ASYNC_TO_LDS_B32 (GVS: SGPR64 base +
    //                 per-lane 32-bit byte offset; LDS dest from VGPR)
    // OOB texel    -> ds_store 0  (implements padding_mode='zeros')
    const uint64_t xbase = (uint64_t)(uintptr_t)x;
    #pragma unroll
    for (int idx = tid; idx < TOTAL; idx += BS * BS) {
        const int ch = idx / TELEMS;
        const int e  = idx - ch * TELEMS;
        const int j  = e / TILE;
        const int i  = e - j * TILE;
        const int r  = gy0 + j;
        const int c  = gx0 + i;
        if ((unsigned)r < (unsigned)IMG_H && (unsigned)c < (unsigned)IMG_W) {
            const uint32_t goff = (uint32_t)((((b * NCH + ch) * IMG_H + r) * IMG_W + c)
                                             * (int)sizeof(float));
            // low 32 bits of a flat pointer into LDS == byte offset for VDST
            const uint32_t loff = (uint32_t)(uintptr_t)&tile[idx];
            asm volatile("global_load_async_to_lds_b32 %0, %1, %2"
                         :: "v"(loff), "v"(goff), "s"(xbase)
                         : "memory");
        } else {
            tile[idx] = 0.0f;
        }
    }
#if __has_builtin(__builtin_amdgcn_s_wait_asynccnt)
    __builtin_amdgcn_s_wait_asynccnt(0);
#else
    asm volatile("s_wait_asynccnt 0x0" ::: "memory");
#endif
    __syncthreads();   // drains DScnt + barrier: all tile writes visible

    // ---- compute: 4 LDS taps, uniform-weight blend, NT coalesced stores ----
    const int ox = bx * BS + tx;
    const int oy = by * BS + ty;
    const int r0 = ty * TILE + tx;
    #pragma unroll
    for (int ch = 0; ch < NCH; ++ch) {
        const float t00 = tile[ch * TELEMS + r0];
        const float t01 = tile[ch * TELEMS + r0 + 1];
        const float t10 = tile[ch * TELEMS + r0 + TILE];
        const float t11 = tile[ch * TELEMS + r0 + TILE + 1];
        const float v = fmaf(t11, w11, fmaf(t10, w10, fmaf(t01, w01, t00 * w00)));
        // write-once stream: non-temporal so it doesn't evict the L2-resident
        // input that neighboring (overlapping) tiles re-read
        __builtin_nontemporal_store(v, &out[((b * NCH + ch) * IMG_H + oy) * IMG_W + ox]);
    }
}

extern "C" void kernel_launch(void* const* d_in, const int* in_sizes, int n_in,
                              void* d_out, int out_size, void* d_ws, size_t ws_size,
                              hipStream_t stream)
{
    (void)n_in; (void)out_size; (void)d_ws; (void)ws_size;
    const float* x    = (const float*)d_in[0];
    const float* flow = (const float*)d_in[1];
    float*       out  = (float*)d_out;

    const int batch = in_sizes[0] / (NCH * IMG_H * IMG_W);   // 32
    dim3 grid(NBX, NBY, batch);
    dim3 block(BS, BS, 1);
    jitter_tile_kernel<<<grid, block, 0, stream>>>(x, flow, out);
}